// DockPointNet_40827959116631
// MI455X (gfx1250) — compile-verified
//
#include <hip/hip_runtime.h>
#include <hip/hip_bf16.h>
#include <math.h>

typedef __attribute__((ext_vector_type(2))) float v2f;
typedef __attribute__((ext_vector_type(8))) float v8f;

#define NNODES 8192
#define NGRAPH 8
#define GSIZE  1024
#define KNN    21          // 20 neighbors + self
#define RAD2   0.04f       // r = 0.2
#define INF_F  __builtin_inff()

// ---------------------------------------------------------------------------
// Generic f32 WMMA GEMM: D[M,Nc] = A[M,K] * B[K,Nc] (+bias, optional relu)
// One wave computes one 16x16 output tile via V_WMMA_F32_16X16X4_F32.
// K is compile-time: main loop has no K guards; single clamped tail step.
// Row/col guards are branch-free (clamped address + select).
// ---------------------------------------------------------------------------
template<int K>
__global__ __launch_bounds__(32)
void gemm_f32_wmma(const float* __restrict__ A, int lda,
                   const float* __restrict__ B, int ldb,
                   float* __restrict__ D, int ldd,
                   int M, int Nc,
                   const float* __restrict__ bias, int relu)
{
    const int rowBase = blockIdx.x * 16;
    const int colBase = blockIdx.y * 16;
    const int lane = threadIdx.x;
    const int l  = lane & 15;
    const int hi = lane >> 4;            // 0: K pair (k,k+1), 1: (k+2,k+3)

    const int row = rowBase + l;
    const int col = colBase + l;
    const bool rowOK = row < M;
    const bool colOK = col < Nc;
    const long aoff = (long)(rowOK ? row : 0) * lda;   // safe A row base
    const int  bcol = colOK ? col : 0;                 // safe B column

    constexpr int KF = K & ~3;                         // full 4-wide steps

    v8f c = {0.f,0.f,0.f,0.f,0.f,0.f,0.f,0.f};
    for (int k0 = 0; k0 < KF; k0 += 4) {
        const int ka = k0 + 2 * hi;
        v2f a, b;
        const float a0 = A[aoff + ka];
        const float a1 = A[aoff + ka + 1];
        const float b0 = B[ka * ldb + bcol];
        const float b1 = B[(ka + 1) * ldb + bcol];
        a.x = rowOK ? a0 : 0.f;  a.y = rowOK ? a1 : 0.f;
        b.x = colOK ? b0 : 0.f;  b.y = colOK ? b1 : 0.f;
        c = __builtin_amdgcn_wmma_f32_16x16x4_f32(false, a, false, b,
                                                  (short)0, c, false, false);
    }
    if constexpr (KF < K) {
        const int ka  = KF + 2 * hi;
        const int i0  = (ka     < K) ? ka     : (K - 1);
        const int i1  = (ka + 1 < K) ? ka + 1 : (K - 1);
        const bool v0 = rowOK && (ka     < K);
        const bool v1 = rowOK && (ka + 1 < K);
        const bool w0 = colOK && (ka     < K);
        const bool w1 = colOK && (ka + 1 < K);
        v2f a, b;
        const float a0 = A[aoff + i0];
        const float a1 = A[aoff + i1];
        const float b0 = B[i0 * ldb + bcol];
        const float b1 = B[i1 * ldb + bcol];
        a.x = v0 ? a0 : 0.f;  a.y = v1 ? a1 : 0.f;
        b.x = w0 ? b0 : 0.f;  b.y = w1 ? b1 : 0.f;
        c = __builtin_amdgcn_wmma_f32_16x16x4_f32(false, a, false, b,
                                                  (short)0, c, false, false);
    }

    if (col < Nc) {
        const float bv = bias ? bias[col] : 0.f;
        for (int r = 0; r < 8; ++r) {
            const int ro = rowBase + hi * 8 + r;
            if (ro < M) {
                float v = c[r] + bv;
                if (relu) v = fmaxf(v, 0.f);
                D[ro * ldd + col] = v;
            }
        }
    }
}

// ---------------------------------------------------------------------------
// Squared norms per node
// ---------------------------------------------------------------------------
template<int DIMS>
__global__ void norms_kernel(const float* __restrict__ X, int lda,
                             float* __restrict__ sq, int n)
{
    int i = blockIdx.x * blockDim.x + threadIdx.x;
    if (i < n) {
        float s = 0.f;
        for (int d = 0; d < DIMS; ++d) { float v = X[i * lda + d]; s += v * v; }
        sq[i] = s;
    }
}

// ---------------------------------------------------------------------------
// Fused kNN: per 16-row block, WMMA Gram vs all 1024 in-graph columns in
// 256-column chunks (LDS), convert to d2, insertion-select top-20 per row.
// One wave per block. Graphs are contiguous blocks of 1024 nodes.
// ---------------------------------------------------------------------------
template<int DIMS>
__global__ __launch_bounds__(32)
void knn_kernel(const float* __restrict__ X, int lda,
                const float* __restrict__ sq,
                int* __restrict__ idxOut, int* __restrict__ mskOut, float r2)
{
    __shared__ float d2s[16 * 256];
    __shared__ float bd[16 * 20];
    __shared__ int   bi[16 * 20];

    const int lane = threadIdx.x;
    const int rowBase = blockIdx.x * 16;
    const int g  = rowBase >> 10;
    const int cb = g << 10;
    const int l  = lane & 15;
    const int hi = lane >> 4;
    const int row = rowBase + l;

    if (lane < 16)
        for (int t = 0; t < 20; ++t) { bd[lane * 20 + t] = INF_F; bi[lane * 20 + t] = 0; }
    __syncthreads();

    float sqr[8];
    for (int r = 0; r < 8; ++r) sqr[r] = sq[rowBase + hi * 8 + r];

    constexpr int KF = DIMS & ~3;

    for (int chunk = 0; chunk < 4; ++chunk) {
        const int cc = cb + chunk * 256;
        for (int tile = 0; tile < 16; ++tile) {
            const int colN = cc + tile * 16;
            const int cn = colN + l;
            v8f c = {0.f,0.f,0.f,0.f,0.f,0.f,0.f,0.f};
            for (int k0 = 0; k0 < KF; k0 += 4) {
                const int ka = k0 + 2 * hi;
                v2f a, b;
                a.x = X[row * lda + ka];     a.y = X[row * lda + ka + 1];
                b.x = X[cn * lda + ka];      b.y = X[cn * lda + ka + 1];
                c = __builtin_amdgcn_wmma_f32_16x16x4_f32(false, a, false, b,
                                                          (short)0, c, false, false);
            }
            if constexpr (KF < DIMS) {
                const int ka = KF + 2 * hi;
                const int i0 = (ka     < DIMS) ? ka     : (DIMS - 1);
                const int i1 = (ka + 1 < DIMS) ? ka + 1 : (DIMS - 1);
                const bool v0 = (ka     < DIMS);
                const bool v1 = (ka + 1 < DIMS);
                v2f a, b;
                const float a0 = X[row * lda + i0];
                const float a1 = X[row * lda + i1];
                const float b0 = X[cn * lda + i0];
                const float b1 = X[cn * lda + i1];
                a.x = v0 ? a0 : 0.f;  a.y = v1 ? a1 : 0.f;
                b.x = v0 ? b0 : 0.f;  b.y = v1 ? b1 : 0.f;
                c = __builtin_amdgcn_wmma_f32_16x16x4_f32(false, a, false, b,
                                                          (short)0, c, false, false);
            }
            const float sqc = sq[cn];
            for (int r = 0; r < 8; ++r) {
                const int ro = rowBase + hi * 8 + r;
                float d = sqr[r] + sqc - 2.f * c[r];
                if (ro == cn) d = INF_F;                 // exclude self
                d2s[(hi * 8 + r) * 256 + tile * 16 + l] = d;
            }
        }
        __syncthreads();
        if (lane < 16) {
            float* bdl = &bd[lane * 20];
            int*   bil = &bi[lane * 20];
            for (int j = 0; j < 256; ++j) {
                const float d = d2s[lane * 256 + j];
                if (d < bdl[19]) {
                    int p = 19;
                    while (p > 0 && bdl[p - 1] > d) {
                        bdl[p] = bdl[p - 1]; bil[p] = bil[p - 1]; --p;
                    }
                    bdl[p] = d; bil[p] = cc + j;
                }
            }
        }
        __syncthreads();
    }

    if (lane < 16) {
        const int ro = rowBase + lane;
        for (int t = 0; t < 20; ++t) {
            idxOut[ro * KNN + t] = bi[lane * 20 + t];
            mskOut[ro * KNN + t] = (bd[lane * 20 + t] <= r2) ? 1 : 0;
        }
        idxOut[ro * KNN + 20] = ro;   // self loop
        mskOut[ro * KNN + 20] = 1;
    }
}

// ---------------------------------------------------------------------------
// Multi-head attention conv message+softmax+max-aggregation (wave per node)
// p: [N, H*C] projected features. agg: [N, H*C] output.
// ---------------------------------------------------------------------------
template<int H, int C>
__global__ __launch_bounds__(32)
void att_kernel(const float* __restrict__ p, const int* __restrict__ idx,
                const int* __restrict__ msk, float* __restrict__ agg)
{
    constexpr int HC = H * C;
    const int i = blockIdx.x;
    const int lane = threadIdx.x;

    __shared__ float s_pi[HC];
    __shared__ float s_s[KNN * C];    // head-summed neighbor feats [k][c]
    __shared__ float s_a[KNN * H];    // scores -> alpha [k][h]
    __shared__ int   s_idx[KNN];
    __shared__ int   s_msk[KNN];

    if (lane < KNN) { s_idx[lane] = idx[i * KNN + lane]; s_msk[lane] = msk[i * KNN + lane]; }
    for (int t = lane; t < HC; t += 32) s_pi[t] = p[(long)i * HC + t];
    __syncthreads();

    // s[k][c] = sum_h p[j][h][c]
    for (int t = lane; t < KNN * C; t += 32) {
        const int k = t / C, c = t % C;
        const float* pj = p + (long)s_idx[k] * HC;
        float s = 0.f;
        for (int h = 0; h < H; ++h) s += pj[h * C + c];
        s_s[k * C + c] = s;
    }
    __syncthreads();

    // scores[k][h] = <p_i[h,:], s[k,:]> / sqrt(C)  (masked -> -inf)
    const float inv = 1.0f / sqrtf((float)C);
    for (int t = lane; t < KNN * H; t += 32) {
        const int k = t % KNN, h = t / KNN;
        float sc;
        if (s_msk[k]) {
            sc = 0.f;
            for (int c = 0; c < C; ++c) sc += s_pi[h * C + c] * s_s[k * C + c];
            sc *= inv;
        } else sc = -INF_F;
        s_a[k * H + h] = sc;
    }
    __syncthreads();

    // softmax over neighborhood per head
    if (lane < H) {
        const int h = lane;
        float m = -INF_F;
        for (int k = 0; k < KNN; ++k) m = fmaxf(m, s_a[k * H + h]);
        float sum = 0.f;
        for (int k = 0; k < KNN; ++k) sum += expf(s_a[k * H + h] - m);
        const float rs = 1.f / sum;
        for (int k = 0; k < KNN; ++k) s_a[k * H + h] = expf(s_a[k * H + h] - m) * rs;
    }
    __syncthreads();

    // agg[h][c] = max over valid k of alpha[k][h] * p[j][h][c]
    for (int t = lane; t < HC; t += 32) {
        const int h = t / C;
        float m = -INF_F;
        for (int k = 0; k < KNN; ++k) {
            if (s_msk[k]) {
                const float v = s_a[k * H + h] * p[(long)s_idx[k] * HC + t];
                m = fmaxf(m, v);
            }
        }
        agg[(long)i * HC + t] = m;
    }
}

// ---------------------------------------------------------------------------
// BatchNorm: per-column batch stats -> scale/shift, then apply in place
// ---------------------------------------------------------------------------
__global__ void bn_stats(const float* __restrict__ h, int M, int C,
                         const float* __restrict__ g, const float* __restrict__ bt,
                         float* __restrict__ scale, float* __restrict__ shift)
{
    const int c = blockIdx.x;
    __shared__ float ssum[256], ssq[256];
    const int t = threadIdx.x;
    float s = 0.f, q = 0.f;
    for (int r = t; r < M; r += 256) { const float v = h[r * C + c]; s += v; q += v * v; }
    ssum[t] = s; ssq[t] = q;
    __syncthreads();
    for (int st = 128; st > 0; st >>= 1) {
        if (t < st) { ssum[t] += ssum[t + st]; ssq[t] += ssq[t + st]; }
        __syncthreads();
    }
    if (t == 0) {
        const float mu  = ssum[0] / (float)M;
        const float var = ssq[0] / (float)M - mu * mu;
        const float sc  = g[c] / sqrtf(var + 1e-5f);
        scale[c] = sc;
        shift[c] = bt[c] - mu * sc;
    }
}

__global__ void bn_apply(float* __restrict__ h, int total, int C,
                         const float* __restrict__ scale, const float* __restrict__ shift)
{
    const int i = blockIdx.x * blockDim.x + threadIdx.x;
    if (i < total) { const int c = i % C; h[i] = h[i] * scale[c] + shift[c]; }
}

// ---------------------------------------------------------------------------
// global_max_pool per graph (graphs are contiguous 1024-row blocks)
// ---------------------------------------------------------------------------
__global__ void pool_kernel(const float* __restrict__ h, float* __restrict__ pooled)
{
    const int b = blockIdx.x;     // graph
    const int c = threadIdx.x;    // 128 channels
    float m = -INF_F;
    for (int r = 0; r < GSIZE; ++r) m = fmaxf(m, h[(b * GSIZE + r) * 128 + c]);
    pooled[b * 128 + c] = m;
}

// ---------------------------------------------------------------------------
// Final head: logits = h3 @ wd + bd ; log_softmax over 2 classes
// ---------------------------------------------------------------------------
__global__ __launch_bounds__(32)
void head_kernel(const float* __restrict__ h3, const float* __restrict__ wd,
                 const float* __restrict__ bdv, float* __restrict__ out)
{
    const int r = threadIdx.x;
    if (r < NGRAPH) {
        float l0 = bdv[0], l1 = bdv[1];
        for (int c = 0; c < 128; ++c) {
            const float v = h3[r * 128 + c];
            l0 += v * wd[c * 2 + 0];
            l1 += v * wd[c * 2 + 1];
        }
        const float m   = fmaxf(l0, l1);
        const float lse = m + logf(expf(l0 - m) + expf(l1 - m));
        out[r * 2 + 0] = l0 - lse;
        out[r * 2 + 1] = l1 - lse;
    }
}

// ---------------------------------------------------------------------------
extern "C" void kernel_launch(void* const* d_in, const int* in_sizes, int n_in,
                              void* d_out, int out_size, void* d_ws, size_t ws_size,
                              hipStream_t stream)
{
    const float* pos = (const float*)d_in[0];
    // d_in[1] = batch (int32): graphs are contiguous 1024-node blocks by construction
    const float* W1  = (const float*)d_in[2];
    const float* lw1 = (const float*)d_in[3];
    const float* lb1 = (const float*)d_in[4];
    const float* W2  = (const float*)d_in[5];
    const float* lw2 = (const float*)d_in[6];
    const float* lb2 = (const float*)d_in[7];
    const float* wa  = (const float*)d_in[8];
    const float* ba  = (const float*)d_in[9];
    const float* ga  = (const float*)d_in[10];
    const float* bta = (const float*)d_in[11];
    const float* wb  = (const float*)d_in[12];
    const float* bb  = (const float*)d_in[13];
    const float* gb  = (const float*)d_in[14];
    const float* btb = (const float*)d_in[15];
    const float* wc  = (const float*)d_in[16];
    const float* bc  = (const float*)d_in[17];
    const float* gc  = (const float*)d_in[18];
    const float* btc = (const float*)d_in[19];
    const float* wd  = (const float*)d_in[20];
    const float* bdv = (const float*)d_in[21];
    float* out = (float*)d_out;

    // ---- workspace layout (bytes, 256-aligned) ----
    char* base = (char*)d_ws;
    size_t o = 0;
    auto alloc = [&](size_t bytes) -> size_t {
        size_t r = o; o += (bytes + 255) & ~(size_t)255; return r;
    };
    const int N = NNODES;
    float* sq    = (float*)(base + alloc((size_t)N * 4));
    int*   idx1  = (int*)  (base + alloc((size_t)N * KNN * 4));
    int*   msk1  = (int*)  (base + alloc((size_t)N * KNN * 4));
    int*   idx2  = (int*)  (base + alloc((size_t)N * KNN * 4));
    int*   msk2  = (int*)  (base + alloc((size_t)N * KNN * 4));
    float* p1    = (float*)(base + alloc((size_t)N * 243 * 4));
    float* agg1  = (float*)(base + alloc((size_t)N * 243 * 4));
    float* feat  = (float*)(base + alloc((size_t)N * 91 * 4));  // [x1 | x2]
    float* p2    = (float*)(base + alloc((size_t)N * 576 * 4));
    float* agg2  = (float*)(base + alloc((size_t)N * 576 * 4));
    float* hbuf  = (float*)(base + alloc((size_t)N * 128 * 4));
    float* scale = (float*)(base + alloc(128 * 4));
    float* shift = (float*)(base + alloc(128 * 4));
    float* pooled= (float*)(base + alloc(NGRAPH * 128 * 4));
    float* h2    = (float*)(base + alloc(NGRAPH * 128 * 4));
    float* h3    = (float*)(base + alloc(NGRAPH * 128 * 4));
    (void)ws_size; (void)in_sizes; (void)n_in; (void)out_size;

    const int TM = N / 16;   // 512 row tiles

    // ---- stage 1: kNN in position space + conv1 ----
    norms_kernel<3><<<(N + 255) / 256, 256, 0, stream>>>(pos, 3, sq, N);
    knn_kernel<3><<<TM, 32, 0, stream>>>(pos, 3, sq, idx1, msk1, RAD2);
    gemm_f32_wmma<3><<<dim3(TM, (243 + 15) / 16), 32, 0, stream>>>(
        pos, 3, W1, 243, p1, 243, N, 243, nullptr, 0);
    att_kernel<9, 27><<<N, 32, 0, stream>>>(p1, idx1, msk1, agg1);
    gemm_f32_wmma<243><<<dim3(TM, (27 + 15) / 16), 32, 0, stream>>>(
        agg1, 243, lw1, 27, feat, 91, N, 27, lb1, 0);           // x1 -> feat[:,0:27]

    // ---- stage 2: kNN in feature space + conv2 ----
    norms_kernel<27><<<(N + 255) / 256, 256, 0, stream>>>(feat, 91, sq, N);
    knn_kernel<27><<<TM, 32, 0, stream>>>(feat, 91, sq, idx2, msk2, RAD2);
    gemm_f32_wmma<27><<<dim3(TM, (576 + 15) / 16), 32, 0, stream>>>(
        feat, 91, W2, 576, p2, 576, N, 576, nullptr, 0);
    att_kernel<9, 64><<<N, 32, 0, stream>>>(p2, idx2, msk2, agg2);
    gemm_f32_wmma<576><<<dim3(TM, (64 + 15) / 16), 32, 0, stream>>>(
        agg2, 576, lw2, 64, feat + 27, 91, N, 64, lb2, 0);      // x2 -> feat[:,27:91]

    // ---- lin1: MLP([91,128]) with batch-stat BN ----
    gemm_f32_wmma<91><<<dim3(TM, 128 / 16), 32, 0, stream>>>(
        feat, 91, wa, 128, hbuf, 128, N, 128, ba, 1);
    bn_stats<<<128, 256, 0, stream>>>(hbuf, N, 128, ga, bta, scale, shift);
    bn_apply<<<(N * 128 + 255) / 256, 256, 0, stream>>>(hbuf, N * 128, 128, scale, shift);

    // ---- global max pool ----
    pool_kernel<<<NGRAPH, 128, 0, stream>>>(hbuf, pooled);

    // ---- mlp: 128->128 (x2) with BN, then head ----
    gemm_f32_wmma<128><<<dim3(1, 128 / 16), 32, 0, stream>>>(
        pooled, 128, wb, 128, h2, 128, NGRAPH, 128, bb, 1);
    bn_stats<<<128, 256, 0, stream>>>(h2, NGRAPH, 128, gb, btb, scale, shift);
    bn_apply<<<(NGRAPH * 128 + 255) / 256, 256, 0, stream>>>(h2, NGRAPH * 128, 128, scale, shift);

    gemm_f32_wmma<128><<<dim3(1, 128 / 16), 32, 0, stream>>>(
        h2, 128, wc, 128, h3, 128, NGRAPH, 128, bc, 1);
    bn_stats<<<128, 256, 0, stream>>>(h3, NGRAPH, 128, gc, btc, scale, shift);
    bn_apply<<<(NGRAPH * 128 + 255) / 256, 256, 0, stream>>>(h3, NGRAPH * 128, 128, scale, shift);

    head_kernel<<<1, 32, 0, stream>>>(h3, wd, bdv, out);
}